// RDetector_37460704756004
// MI455X (gfx1250) — compile-verified
//
#include <hip/hip_runtime.h>
#include <math.h>

// RDetector: conv1d-embed + 4x transformer encoder + MLP decoder.
// Attention uses V_WMMA_F32_16X16X4_F32 (head_dim=4 == WMMA K) in a
// flash-attention loop, one wave32 per 16-query tile per (batch, head).
// Softmax row-max uses DPP8 + permlane16 (pure VALU); softmax row-sum is
// accumulated for free in column 4 of the PV WMMA accumulator.

#define E  20
#define NH 5
#define HD 4
#define FF 120
#define DEC 40
#define KW 50
#define PADW 25
#define BB 2
#define LL 5000

typedef __attribute__((ext_vector_type(2))) float v2f;
typedef __attribute__((ext_vector_type(8))) float v8f;

__host__ __device__ constexpr unsigned dpp8sel(int a, int b, int c, int d,
                                               int e, int f, int g, int h) {
  return (unsigned)(a | (b << 3) | (c << 6) | (d << 9) | (e << 12) |
                    (f << 15) | (g << 18) | (h << 21));
}

#define DPP8_XOR1 dpp8sel(1, 0, 3, 2, 5, 4, 7, 6)
#define DPP8_XOR2 dpp8sel(2, 3, 0, 1, 6, 7, 4, 5)
#define DPP8_XOR4 dpp8sel(4, 5, 6, 7, 0, 1, 2, 3)

// max across a 16-lane row (lanes 0-15 / 16-31 reduced independently)
__device__ inline float rowmax16(float x) {
  float t;
  t = __builtin_bit_cast(float, __builtin_amdgcn_mov_dpp8(
          __builtin_bit_cast(int, x), DPP8_XOR1));
  x = fmaxf(x, t);
  t = __builtin_bit_cast(float, __builtin_amdgcn_mov_dpp8(
          __builtin_bit_cast(int, x), DPP8_XOR2));
  x = fmaxf(x, t);
  t = __builtin_bit_cast(float, __builtin_amdgcn_mov_dpp8(
          __builtin_bit_cast(int, x), DPP8_XOR4));
  x = fmaxf(x, t);
  // xor8 within each 16-lane row: v_permlane16_b32
  int xi = __builtin_bit_cast(int, x);
  int p = __builtin_amdgcn_permlane16(xi, xi, (int)0xFEDCBA98, 0x76543210,
                                      false, false);
  x = fmaxf(x, __builtin_bit_cast(float, p));
  return x;
}

// ---------------- conv embed + positional encoding -> h (B,L,E) -------------
__global__ __launch_bounds__(256) void conv_pe_kernel(const float* __restrict__ x,
                                                      const float* __restrict__ w,
                                                      float* __restrict__ h) {
  int idx = blockIdx.x * blockDim.x + threadIdx.x;
  if (idx >= BB * E * LL) return;
  int l = idx % LL;
  int e = (idx / LL) % E;
  int b = idx / (LL * E);
  const float* xb = x + b * LL;
  const float* we = w + e * KW;
  float acc = 0.f;
  for (int k = 0; k < KW; ++k) {
    int xi = l + k - PADW;
    if (xi >= 0 && xi < LL) acc += xb[xi] * we[k];
  }
  // pe[r,l] = (r even ? sin : cos)(l * 10000^(-r/10))  (both parities collapse
  // to exponent 2r/E in the reference's loop)
  float inv = powf(10000.f, -0.1f * (float)e);
  float ang = (float)l * inv;
  float pe = (e & 1) ? cosf(ang) : sinf(ang);
  h[(b * LL + l) * E + e] = acc + pe;
}

// ---------------- per-head QKV projection (D x D weights) -------------------
__global__ __launch_bounds__(256) void qkv_kernel(const float* __restrict__ h,
                                                  const float* __restrict__ Wq,
                                                  const float* __restrict__ Wk,
                                                  const float* __restrict__ Wv,
                                                  float* __restrict__ Q,
                                                  float* __restrict__ K,
                                                  float* __restrict__ V) {
  int idx = blockIdx.x * blockDim.x + threadIdx.x;
  if (idx >= BB * NH * LL) return;
  int l = idx % LL;
  int hh = (idx / LL) % NH;
  int b = idx / (LL * NH);
  const float* s = h + (b * LL + l) * E + hh * HD;
  float s0 = s[0], s1 = s[1], s2 = s[2], s3 = s[3];
  int o = ((b * NH + hh) * LL + l) * HD;
#pragma unroll
  for (int d = 0; d < HD; ++d) {
    Q[o + d] = Wq[d*4+0]*s0 + Wq[d*4+1]*s1 + Wq[d*4+2]*s2 + Wq[d*4+3]*s3;
    K[o + d] = Wk[d*4+0]*s0 + Wk[d*4+1]*s1 + Wk[d*4+2]*s2 + Wk[d*4+3]*s3;
    V[o + d] = Wv[d*4+0]*s0 + Wv[d*4+1]*s1 + Wv[d*4+2]*s2 + Wv[d*4+3]*s3;
  }
}

// ---------------- flash attention: one wave32 per 16-query tile -------------
__global__ __launch_bounds__(32) void attn_kernel(const float* __restrict__ Q,
                                                  const float* __restrict__ K,
                                                  const float* __restrict__ V,
                                                  float* __restrict__ O) {
  __shared__ float sP[16][18];
  const int lane = threadIdx.x;
  const int mh = lane & 15;   // M row (A) / N col (B,C,D)
  const int kp = lane >> 4;   // lane-half: selects K pair / row group
  const int bh = blockIdx.y;  // (b, head)
  const int q0 = blockIdx.x * 16;
  const float* Qb = Q + bh * LL * HD;
  const float* Kb = K + bh * LL * HD;
  const float* Vb = V + bh * LL * HD;
  const float scale = 0.22360679775f;  // 1/sqrt(E)   (reference uses E, not D)

  // A operand: Q tile (16x4), pre-scaled. lane mh holds K = 2*kp, 2*kp+1.
  v2f a = {0.f, 0.f};
  {
    int q = q0 + mh;
    if (q < LL) {
      a.x = Qb[q * HD + 2 * kp + 0] * scale;
      a.y = Qb[q * HD + 2 * kp + 1] * scale;
    }
  }

  v8f acc = {0.f, 0.f, 0.f, 0.f, 0.f, 0.f, 0.f, 0.f};
  float row_m[8];
#pragma unroll
  for (int r = 0; r < 8; ++r) row_m[r] = -1.0e30f;

  const int nkt = (LL + 15) / 16;
  for (int kt = 0; kt < nkt; ++kt) {
    const int k0 = kt * 16;
    // B operand: K-tile^T (4x16). lane mh = key column, K(dhead) = 2*kp+p.
    v2f bm = {0.f, 0.f};
    const int key = k0 + mh;
    const bool colvalid = (key < LL);
    if (colvalid) {
      bm.x = Kb[key * HD + 2 * kp + 0];
      bm.y = Kb[key * HD + 2 * kp + 1];
    }
    v8f zero = {0.f, 0.f, 0.f, 0.f, 0.f, 0.f, 0.f, 0.f};
    // S(16x16) = Q(16x4) x K^T(4x16)
    v8f s = __builtin_amdgcn_wmma_f32_16x16x4_f32(false, a, false, bm,
                                                  (short)0, zero, false, false);
    // online softmax; row M = r + 8*kp spread across 16 lanes (col = mh)
#pragma unroll
    for (int r = 0; r < 8; ++r) {
      float sv = colvalid ? s[r] : -3.0e30f;
      float mx = rowmax16(sv);                 // pure-VALU 16-lane max
      float mnew = fmaxf(row_m[r], mx);
      float alpha = __expf(row_m[r] - mnew);
      row_m[r] = mnew;
      float p = __expf(sv - mnew);
      acc[r] *= alpha;                         // rescales O cols AND sum col
      sP[r + 8 * kp][mh] = p;                  // stage P (D-layout->row-major)
    }
    __syncthreads();
    // O(16x16) += P(16x16) x V(16x4 | ones): 4 chained K=4 WMMAs.
    // B col 0..3 = V head dims, col 4 = 1.0 -> acc[:,4] = running softmax sum.
#pragma unroll
    for (int j = 0; j < 4; ++j) {
      v2f pa;
      pa.x = sP[mh][4 * j + 2 * kp + 0];  // A: row mh, K = 2*kp+p of chunk j
      pa.y = sP[mh][4 * j + 2 * kp + 1];
      v2f vb = {0.f, 0.f};
      const int krow0 = k0 + 4 * j + 2 * kp;
      if (mh < HD) {  // B: row = local k, col = head-dim
        if (krow0 < LL)     vb.x = Vb[krow0 * HD + mh];
        if (krow0 + 1 < LL) vb.y = Vb[(krow0 + 1) * HD + mh];
      } else if (mh == HD) {  // ones column: row-sum accumulator
        vb.x = 1.f;
        vb.y = 1.f;
      }
      acc = __builtin_amdgcn_wmma_f32_16x16x4_f32(false, pa, false, vb,
                                                  (short)0, acc, false, false);
    }
    __syncthreads();
  }

  // store O rows (head dims in cols 0..3; softmax sum in col 4), (B,L,E)
  const int b = bh / NH, hh = bh % NH;
#pragma unroll
  for (int r = 0; r < 8; ++r) {
    float sum = __shfl(acc[r], HD, 16);  // broadcast col-4 value across row
    int q = q0 + r + 8 * kp;
    if (mh < HD && q < LL) O[(b * LL + q) * E + hh * HD + mh] = acc[r] / sum;
  }
}

// ---------------- out-projection + residual + layernorm A ------------------
__global__ __launch_bounds__(128) void projln_kernel(const float* __restrict__ O,
                                                     const float* __restrict__ h,
                                                     const float* __restrict__ Wc,
                                                     const float* __restrict__ bc,
                                                     const float* __restrict__ g,
                                                     const float* __restrict__ bt,
                                                     float* __restrict__ h1) {
  int idx = blockIdx.x * blockDim.x + threadIdx.x;
  if (idx >= BB * LL) return;
  const float* orow = O + idx * E;
  const float* hrow = h + idx * E;
  float ov[E], r[E];
#pragma unroll
  for (int e = 0; e < E; ++e) ov[e] = orow[e];
  float mu = 0.f;
#pragma unroll
  for (int e = 0; e < E; ++e) {
    float acc = bc[e];
#pragma unroll
    for (int e2 = 0; e2 < E; ++e2) acc += Wc[e * E + e2] * ov[e2];
    r[e] = acc + hrow[e];
    mu += r[e];
  }
  mu *= (1.f / E);
  float var = 0.f;
#pragma unroll
  for (int e = 0; e < E; ++e) { float d = r[e] - mu; var += d * d; }
  var *= (1.f / E);
  float rs = rsqrtf(var + 1e-5f);
#pragma unroll
  for (int e = 0; e < E; ++e) h1[idx * E + e] = (r[e] - mu) * rs * g[e] + bt[e];
}

// ---------------- feed-forward + residual + layernorm B --------------------
__global__ __launch_bounds__(128) void ffln_kernel(const float* __restrict__ h1,
                                                   const float* __restrict__ W1,
                                                   const float* __restrict__ b1,
                                                   const float* __restrict__ W2,
                                                   const float* __restrict__ b2,
                                                   const float* __restrict__ g,
                                                   const float* __restrict__ bt,
                                                   float* __restrict__ h) {
  int idx = blockIdx.x * blockDim.x + threadIdx.x;
  if (idx >= BB * LL) return;
  float x[E], out[E], r[E];
#pragma unroll
  for (int e = 0; e < E; ++e) { x[e] = h1[idx * E + e]; out[e] = b2[e]; }
  for (int j = 0; j < FF; ++j) {
    float acc = b1[j];
#pragma unroll
    for (int e = 0; e < E; ++e) acc += W1[j * E + e] * x[e];
    acc = fmaxf(acc, 0.f);
#pragma unroll
    for (int e = 0; e < E; ++e) out[e] += W2[e * FF + j] * acc;
  }
  float mu = 0.f;
#pragma unroll
  for (int e = 0; e < E; ++e) { r[e] = out[e] + x[e]; mu += r[e]; }
  mu *= (1.f / E);
  float var = 0.f;
#pragma unroll
  for (int e = 0; e < E; ++e) { float d = r[e] - mu; var += d * d; }
  var *= (1.f / E);
  float rs = rsqrtf(var + 1e-5f);
#pragma unroll
  for (int e = 0; e < E; ++e) h[idx * E + e] = (r[e] - mu) * rs * g[e] + bt[e];
}

// ---------------- decoder MLP + sigmoid -------------------------------------
__global__ __launch_bounds__(128) void dec_kernel(const float* __restrict__ h,
    const float* __restrict__ f1w, const float* __restrict__ f1b,
    const float* __restrict__ f2w, const float* __restrict__ f2b,
    const float* __restrict__ f3w, const float* __restrict__ f3b,
    const float* __restrict__ f4w, const float* __restrict__ f4b,
    float* __restrict__ out) {
  int idx = blockIdx.x * blockDim.x + threadIdx.x;
  if (idx >= BB * LL) return;
  float x[E];
#pragma unroll
  for (int e = 0; e < E; ++e) x[e] = h[idx * E + e];
  float d1[DEC];
#pragma unroll
  for (int j = 0; j < DEC; ++j) {
    float acc = f1b[j];
#pragma unroll
    for (int e = 0; e < E; ++e) acc += f1w[j * E + e] * x[e];
    d1[j] = fmaxf(acc, 0.f);
  }
  float d2[DEC];
#pragma unroll
  for (int j = 0; j < DEC; ++j) {
    float acc = f2b[j];
#pragma unroll
    for (int t = 0; t < DEC; ++t) acc += f2w[j * DEC + t] * d1[t];
    d2[j] = fmaxf(acc, 0.f);
  }
  float d3[E];
#pragma unroll
  for (int e = 0; e < E; ++e) {
    float acc = f3b[e];
#pragma unroll
    for (int t = 0; t < DEC; ++t) acc += f3w[e * DEC + t] * d2[t];
    d3[e] = fmaxf(acc, 0.f);
  }
  float z = f4b[0];
#pragma unroll
  for (int e = 0; e < E; ++e) z += f4w[e] * d3[e];
  out[idx] = 1.f / (1.f + expf(-z));
}

extern "C" void kernel_launch(void* const* d_in, const int* in_sizes, int n_in,
                              void* d_out, int out_size, void* d_ws, size_t ws_size,
                              hipStream_t stream) {
  const float* x      = (const float*)d_in[0];
  const float* conv_w = (const float*)d_in[1];
  const float* Wv     = (const float*)d_in[2];
  const float* Wk     = (const float*)d_in[3];
  const float* Wq     = (const float*)d_in[4];
  const float* Wc     = (const float*)d_in[5];
  const float* bc     = (const float*)d_in[6];
  const float* lnA_g  = (const float*)d_in[7];
  const float* lnA_b  = (const float*)d_in[8];
  const float* W1     = (const float*)d_in[9];
  const float* b1     = (const float*)d_in[10];
  const float* W2     = (const float*)d_in[11];
  const float* b2     = (const float*)d_in[12];
  const float* lnB_g  = (const float*)d_in[13];
  const float* lnB_b  = (const float*)d_in[14];
  const float* f1w    = (const float*)d_in[15];
  const float* f1b    = (const float*)d_in[16];
  const float* f2w    = (const float*)d_in[17];
  const float* f2b    = (const float*)d_in[18];
  const float* f3w    = (const float*)d_in[19];
  const float* f3b    = (const float*)d_in[20];
  const float* f4w    = (const float*)d_in[21];
  const float* f4b    = (const float*)d_in[22];
  float* outp = (float*)d_out;

  float* ws = (float*)d_ws;
  const int TOK = BB * LL * E;  // 200000
  float* hb = ws;
  float* h1 = ws + 1 * TOK;
  float* Qb = ws + 2 * TOK;
  float* Kb = ws + 3 * TOK;
  float* Vb = ws + 4 * TOK;
  float* Ob = ws + 5 * TOK;

  conv_pe_kernel<<<(BB * E * LL + 255) / 256, 256, 0, stream>>>(x, conv_w, hb);
  for (int i = 0; i < 4; ++i) {
    qkv_kernel<<<(BB * NH * LL + 255) / 256, 256, 0, stream>>>(
        hb, Wq + i * HD * HD, Wk + i * HD * HD, Wv + i * HD * HD, Qb, Kb, Vb);
    dim3 ag((LL + 15) / 16, BB * NH);
    attn_kernel<<<ag, 32, 0, stream>>>(Qb, Kb, Vb, Ob);
    projln_kernel<<<(BB * LL + 127) / 128, 128, 0, stream>>>(
        Ob, hb, Wc + i * E * E, bc + i * E, lnA_g + i * E, lnA_b + i * E, h1);
    ffln_kernel<<<(BB * LL + 127) / 128, 128, 0, stream>>>(
        h1, W1 + i * FF * E, b1 + i * FF, W2 + i * E * FF, b2 + i * E,
        lnB_g + i * E, lnB_b + i * E, hb);
  }
  dec_kernel<<<(BB * LL + 127) / 128, 128, 0, stream>>>(
      hb, f1w, f1b, f2w, f2b, f3w, f3b, f4w, f4b, outp);
}